// RelationalReasoning_54348516163955
// MI455X (gfx1250) — compile-verified
//
#include <hip/hip_runtime.h>
#include <hip/hip_bf16.h>

typedef __attribute__((ext_vector_type(2))) float v2f;
typedef __attribute__((ext_vector_type(8))) float v8f;

#define BSZ   256
#define KIMG  16
#define FT    512
#define HID   256
#define NPAIR 120
#define NPOS  (NPAIR * BSZ)   // 30720
#define NROWS (2 * NPOS)      // 61440
#define EPSV  1e-5f
#define SLOPE 0.01f

// workspace layout (float offsets)
#define WS_U     0
#define WS_V     (WS_U + 4096 * HID)      // 1,048,576
#define WS_PSUM  (WS_V + 4096 * HID)      // 2,097,152
#define WS_PSQ   (WS_PSUM + 256 * HID)
#define WS_SCALE (WS_PSQ + 256 * HID)
#define WS_SHIFT (WS_SCALE + HID)
#define WS_LOSS  (WS_SHIFT + HID)         // 7680 entries

// ---------------------------------------------------------------------------
// Pass 1: U = X @ W1[:512,:], V = X @ W1[512:,:]  (X = cnn_output as 4096x512)
// fp32 WMMA 16x16x4. One wave computes a 16(M) x 64(N) tile (4 accumulators,
// A-fragment reused). grid: 2 matrices * 256 Mtiles * 4 Ngroups = 2048 waves.
// ---------------------------------------------------------------------------
__global__ __launch_bounds__(256) void gemm_uv(const float* __restrict__ X,
                                               const float* __restrict__ W1,
                                               float* __restrict__ ws) {
  const int lane = threadIdx.x & 31;
  const int wave = threadIdx.x >> 5;
  const int gw   = blockIdx.x * 8 + wave;
  const int mat  = gw >> 10;          // 0 -> U (W1 top), 1 -> V (W1 bottom)
  const int rem  = gw & 1023;
  const int m0   = (rem >> 2) << 4;   // M tile base (0..4080)
  const int n0   = (rem & 3) << 6;    // N group base (0,64,128,192)

  const float* W   = W1 + (mat ? (size_t)FT * HID : 0);
  float*       OUT = ws + (mat ? WS_V : WS_U);

  const int am = lane & 15;           // A: row M within tile
  const int kh = lane >> 4;           // A/B: K half (0 -> K0/K1, 1 -> K2/K3)
  const float* xrow = X + (size_t)(m0 + am) * FT;

  v8f c0 = {}, c1 = {}, c2 = {}, c3 = {};

#pragma unroll 2
  for (int kk = 0; kk < FT; kk += 4) {
    const int ka = kk + 2 * kh;
    v2f a = *(const v2f*)(xrow + ka);              // A[M=am, K=ka..ka+1]
    const float* wp0 = W + (size_t)ka * HID + am;  // B row K=ka, col base
    const float* wp1 = wp0 + HID;                  // B row K=ka+1
    v2f b0, b1v, b2v, b3v;
    b0.x  = wp0[n0 +  0]; b0.y  = wp1[n0 +  0];
    b1v.x = wp0[n0 + 16]; b1v.y = wp1[n0 + 16];
    b2v.x = wp0[n0 + 32]; b2v.y = wp1[n0 + 32];
    b3v.x = wp0[n0 + 48]; b3v.y = wp1[n0 + 48];
    c0 = __builtin_amdgcn_wmma_f32_16x16x4_f32(false, a, false, b0,  (short)0, c0, false, false);
    c1 = __builtin_amdgcn_wmma_f32_16x16x4_f32(false, a, false, b1v, (short)0, c1, false, false);
    c2 = __builtin_amdgcn_wmma_f32_16x16x4_f32(false, a, false, b2v, (short)0, c2, false, false);
    c3 = __builtin_amdgcn_wmma_f32_16x16x4_f32(false, a, false, b3v, (short)0, c3, false, false);
  }

  // C/D layout: lanes 0-15 -> M=r, lanes 16-31 -> M=8+r; N = n_tile + (lane&15)
  const int nlo = lane & 15;
  const int mhi = (lane >> 4) * 8;
#pragma unroll
  for (int r = 0; r < 8; ++r) {
    float* orow = OUT + (size_t)(m0 + mhi + r) * HID + n0 + nlo;
    orow[ 0] = c0[r];
    orow[16] = c1[r];
    orow[32] = c2[r];
    orow[48] = c3[r];
  }
}

// ---------------------------------------------------------------------------
// Row decode: row -> (gi, gj) indices into the 4096-row U/V matrices.
// ---------------------------------------------------------------------------
__device__ __forceinline__ void decode_row(int row, int& gi, int& gj) {
  int r = row;
  int isneg = 0;
  if (r >= NPOS) { r -= NPOS; isneg = 1; }
  const int pidx = r >> 8;      // pair index (0..119)
  const int b    = r & 255;     // batch index
  int i = 0, remp = pidx;
  while (remp >= 15 - i) { remp -= 15 - i; ++i; }   // triu_indices(16, k=1)
  const int j = i + 1 + remp;
  gi = b * KIMG + i;
  int bj = b;
  if (isneg) bj = (b - (pidx + 1) + BSZ) & (BSZ - 1);  // shift = pidx%255 + 1
  gj = bj * KIMG + j;
}

// ---------------------------------------------------------------------------
// Pass 2: per-block partial sum / sumsq of h over 240 rows (thread = column).
// ---------------------------------------------------------------------------
#define ROWS_PER_BLK 240
__global__ __launch_bounds__(256) void stats_pass(const float* __restrict__ U,
                                                  const float* __restrict__ V,
                                                  const float* __restrict__ b1,
                                                  float* __restrict__ psum,
                                                  float* __restrict__ psq) {
  const int c  = threadIdx.x;
  const float bc = b1[c];
  float s = 0.f, q = 0.f;
  const int r0 = blockIdx.x * ROWS_PER_BLK;
  for (int r = r0; r < r0 + ROWS_PER_BLK; ++r) {
    int gi, gj; decode_row(r, gi, gj);
    const float h = U[(size_t)gi * HID + c] + V[(size_t)gj * HID + c] + bc;
    s += h; q += h * h;
  }
  psum[(size_t)blockIdx.x * HID + c] = s;
  psq [(size_t)blockIdx.x * HID + c] = q;
}

// ---------------------------------------------------------------------------
// Pass 3: fold partials -> per-column affine (scale, shift) for BN.
// ---------------------------------------------------------------------------
__global__ void finalize_stats(const float* __restrict__ psum,
                               const float* __restrict__ psq,
                               const float* __restrict__ gamma,
                               const float* __restrict__ beta,
                               float* __restrict__ scale,
                               float* __restrict__ shiftc) {
  const int c = threadIdx.x;
  float s = 0.f, q = 0.f;
  for (int b = 0; b < 256; ++b) {   // fixed order -> deterministic
    s += psum[(size_t)b * HID + c];
    q += psq [(size_t)b * HID + c];
  }
  const float inv  = 1.0f / (float)NROWS;
  const float mean = s * inv;
  const float var  = q * inv - mean * mean;
  const float sc   = gamma[c] * rsqrtf(var + EPSV);
  scale[c]  = sc;
  shiftc[c] = beta[c] - mean * sc;
}

// ---------------------------------------------------------------------------
// Pass 4: one wave per row: BN + leaky-ReLU + dot(W2) -> score, 9 thresholds,
// per-block loss partial.  d_out = [loss | scores(61440) | correct(9*61440)]
// ---------------------------------------------------------------------------
__global__ __launch_bounds__(256) void score_pass(const float* __restrict__ U,
                                                  const float* __restrict__ V,
                                                  const float* __restrict__ b1,
                                                  const float* __restrict__ scale,
                                                  const float* __restrict__ shiftc,
                                                  const float* __restrict__ W2,
                                                  const float* __restrict__ b2,
                                                  float* __restrict__ out,
                                                  float* __restrict__ loss_part) {
  __shared__ float ls[8];
  const int lane = threadIdx.x & 31;
  const int wave = threadIdx.x >> 5;
  const int row  = blockIdx.x * 8 + wave;
  int gi, gj; decode_row(row, gi, gj);
  const float* urow = U + (size_t)gi * HID;
  const float* vrow = V + (size_t)gj * HID;

  float acc = 0.f;
#pragma unroll
  for (int k = 0; k < 8; ++k) {
    const int c = lane + 32 * k;
    float h  = urow[c] + vrow[c] + b1[c];
    float hn = h * scale[c] + shiftc[c];
    hn = (hn >= 0.f) ? hn : SLOPE * hn;
    acc += hn * W2[c];
  }
#pragma unroll
  for (int off = 16; off > 0; off >>= 1) acc += __shfl_xor(acc, off, 32);

  if (lane == 0) {
    const float sc  = acc + b2[0];
    out[1 + row] = sc;
    const float tgt = (row < NPOS) ? 1.f : 0.f;
    ls[wave] = fmaxf(sc, 0.f) - sc * tgt + log1pf(expf(-fabsf(sc)));
#pragma unroll
    for (int t = 1; t <= 9; ++t) {
      const float pred = (sc > (float)t * 0.1f) ? 1.f : 0.f;
      out[1 + NROWS + (size_t)(t - 1) * NROWS + row] = (pred == tgt) ? 1.f : 0.f;
    }
  }
  __syncthreads();
  if (threadIdx.x == 0) {
    float t = 0.f;
#pragma unroll
    for (int w = 0; w < 8; ++w) t += ls[w];
    loss_part[blockIdx.x] = t;
  }
}

// ---------------------------------------------------------------------------
// Pass 5: deterministic fixed-order loss reduction -> d_out[0]
// ---------------------------------------------------------------------------
__global__ void loss_reduce(const float* __restrict__ loss_part,
                            float* __restrict__ out) {
  __shared__ float sm[256];
  const int t = threadIdx.x;
  float s = 0.f;
  for (int i = t; i < NROWS / 8; i += 256) s += loss_part[i];
  sm[t] = s;
  __syncthreads();
  for (int off = 128; off > 0; off >>= 1) {
    if (t < off) sm[t] += sm[t + off];
    __syncthreads();
  }
  if (t == 0) out[0] = sm[0] / (float)NROWS;
}

extern "C" void kernel_launch(void* const* d_in, const int* in_sizes, int n_in,
                              void* d_out, int out_size, void* d_ws, size_t ws_size,
                              hipStream_t stream) {
  const float* X     = (const float*)d_in[0];  // cnn_output (256,16,512)
  const float* W1    = (const float*)d_in[1];  // (1024,256)
  const float* b1    = (const float*)d_in[2];  // (256,)
  const float* gamma = (const float*)d_in[3];  // (256,)
  const float* beta  = (const float*)d_in[4];  // (256,)
  const float* W2    = (const float*)d_in[5];  // (256,1)
  const float* b2    = (const float*)d_in[6];  // (1,)
  float* out = (float*)d_out;
  float* ws  = (float*)d_ws;                   // needs ~9 MB

  gemm_uv<<<256, 256, 0, stream>>>(X, W1, ws);
  stats_pass<<<NROWS / ROWS_PER_BLK, 256, 0, stream>>>(
      ws + WS_U, ws + WS_V, b1, ws + WS_PSUM, ws + WS_PSQ);
  finalize_stats<<<1, 256, 0, stream>>>(
      ws + WS_PSUM, ws + WS_PSQ, gamma, beta, ws + WS_SCALE, ws + WS_SHIFT);
  score_pass<<<NROWS / 8, 256, 0, stream>>>(
      ws + WS_U, ws + WS_V, b1, ws + WS_SCALE, ws + WS_SHIFT, W2, b2,
      out, ws + WS_LOSS);
  loss_reduce<<<1, 256, 0, stream>>>(ws + WS_LOSS, out);
}